// Rule_58643483460147
// MI455X (gfx1250) — compile-verified
//
#include <hip/hip_runtime.h>
#include <cstdint>

// ---------------- problem constants ----------------
constexpr int kH = 1024;
constexpr int kW = 1024;
constexpr int kRadius = 3;
constexpr int kRK = 2 * kRadius + 1;      // 7
constexpr int kPlane = kH * kW;
constexpr float kTraceMem = 0.995f;

// ---------------- tiling ----------------
constexpr int TILE = 32;                   // 32x32 pixels per block
constexpr int HALO_W = TILE + 2 * kRadius; // 38
constexpr int LDS_STRIDE = 40;             // padded row stride (floats)
constexpr int BLOCK = 256;                 // 8 waves (wave32)

typedef float v4f __attribute__((ext_vector_type(4)));

// gfx1250 async global->LDS path (probe via __has_builtin; fallback = load + ds_store)
#if defined(__gfx1250__) && __has_builtin(__builtin_amdgcn_global_load_async_to_lds_b32) && __has_builtin(__builtin_amdgcn_s_wait_asynccnt)
#define USE_ASYNC_LDS 1
#else
#define USE_ASYNC_LDS 0
#endif

#if USE_ASYNC_LDS
typedef __attribute__((address_space(1))) int* gptr_i32;
typedef __attribute__((address_space(3))) int* lptr_i32;
#endif

__global__ __launch_bounds__(BLOCK) void ising_stencil_kernel(
    const float* __restrict__ x,      // (3,H,W): s, tr, b
    const float* __restrict__ J,      // (49,H,W)
    const float* __restrict__ rflip,  // (1,1,H,W)
    const float* __restrict__ rdrop,  // (H,W)
    float* __restrict__ out)          // (3,H,W)
{
  __shared__ float tile[HALO_W * LDS_STRIDE];

  const int tid   = threadIdx.x;
  const int nTx   = kW / TILE;                       // tiles per row
  const int tileC = (blockIdx.x % nTx) * TILE;
  const int tileR = (blockIdx.x / nTx) * TILE;

  const float* s  = x;                // plane 0
  const float* tr = x + kPlane;       // plane 1
  const float* b  = x + 2 * kPlane;   // plane 2

  // ---- cooperative halo tile load (periodic wrap), async to LDS if available ----
  for (int idx = tid; idx < HALO_W * HALO_W; idx += BLOCK) {
    const int r  = idx / HALO_W;
    const int c  = idx - r * HALO_W;
    const int gr = (tileR + r - kRadius) & (kH - 1);
    const int gc = (tileC + c - kRadius) & (kW - 1);
#if USE_ASYNC_LDS
    __builtin_amdgcn_global_load_async_to_lds_b32(
        (gptr_i32)(s + gr * kW + gc),
        (lptr_i32)&tile[r * LDS_STRIDE + c],
        0, 0);
#else
    tile[r * LDS_STRIDE + c] = s[gr * kW + gc];
#endif
  }
#if USE_ASYNC_LDS
  __builtin_amdgcn_s_wait_asynccnt(0);
#endif
  __syncthreads();

  // ---- each thread computes 4 consecutive pixels in x ----
  const int tx = tid & 7;    // column group: cols 4*tx .. 4*tx+3
  const int ty = tid >> 3;   // row 0..31
  const int gi = tileR + ty;
  const int gj = tileC + tx * 4;

  float acc0 = 0.f, acc1 = 0.f, acc2 = 0.f, acc3 = 0.f;
  const float* Jp = J + (size_t)gi * kW + gj;

#pragma unroll
  for (int oi = 0; oi < kRK; ++oi) {
    // s row slice needed for this oi: halo cols [4*tx .. 4*tx+9]
    float srow[10];
#pragma unroll
    for (int m = 0; m < 10; ++m)
      srow[m] = tile[(ty + oi) * LDS_STRIDE + tx * 4 + m];
#pragma unroll
    for (int oj = 0; oj < kRK; ++oj) {
      // J is streamed exactly once -> non-temporal so it doesn't wash L2
      const v4f w = __builtin_nontemporal_load(
          (const v4f*)(Jp + (size_t)(oi * kRK + oj) * kPlane));
      acc0 = fmaf(srow[oj + 0], w.x, acc0);
      acc1 = fmaf(srow[oj + 1], w.y, acc1);
      acc2 = fmaf(srow[oj + 2], w.z, acc2);
      acc3 = fmaf(srow[oj + 3], w.w, acc3);
    }
  }

  // center s values straight from LDS
  float sv[4];
#pragma unroll
  for (int m = 0; m < 4; ++m)
    sv[m] = tile[(ty + kRadius) * LDS_STRIDE + tx * 4 + kRadius + m];

  const int pix = gi * kW + gj;
  const v4f tr4 = *(const v4f*)(tr + pix);
  const v4f b4  = *(const v4f*)(b + pix);
  const v4f rf4 = *(const v4f*)(rflip + pix);
  const v4f rd4 = *(const v4f*)(rdrop + pix);

  const float acc[4] = {acc0, acc1, acc2, acc3};
  v4f outS, outT;
#pragma unroll
  for (int m = 0; m < 4; ++m) {
    const float de = 2.f * sv[m] * acc[m];
    float p;
    if (de <= 0.f) {
      p = 1.f;
    } else {
      p = expf(-de * b4[m]);
    }
    const bool doFlip = (rf4[m] < p) && (rd4[m] > 0.5f);
    const float flip  = doFlip ? -1.f : 1.f;
    outS[m] = sv[m] * flip;
    outT[m] = fmaf(kTraceMem, tr4[m], (1.f - kTraceMem) * sv[m]);
  }

  *(v4f*)(out + pix)              = outS;
  *(v4f*)(out + kPlane + pix)     = outT;
  *(v4f*)(out + 2 * kPlane + pix) = b4;   // b passes through
}

extern "C" void kernel_launch(void* const* d_in, const int* in_sizes, int n_in,
                              void* d_out, int out_size, void* d_ws, size_t ws_size,
                              hipStream_t stream) {
  const float* x     = (const float*)d_in[0];  // (3,H,W)
  const float* J     = (const float*)d_in[1];  // (49,H,W)
  const float* rflip = (const float*)d_in[2];  // (H,W)
  const float* rdrop = (const float*)d_in[3];  // (H,W)
  float* out = (float*)d_out;                  // (3,H,W)

  const int nBlocks = (kH / TILE) * (kW / TILE);  // 1024
  ising_stencil_kernel<<<nBlocks, BLOCK, 0, stream>>>(x, J, rflip, rdrop, out);
}